// SoftMSMLoss_17678085390885
// MI455X (gfx1250) — compile-verified
//
#include <hip/hip_runtime.h>
#include <stdint.h>

#define TLEN 512   // T == U == 512
#define BATCH 64

// ---------------------------------------------------------------------------
// Soft-MSM math (c = 1.0, gamma = 1.0, eps = 1e-9), fast transcendentals.
// softmin2(p,q) = -logaddexp(-p,-q) = min(p,q) - log1p(exp(-|p-q|))
// ---------------------------------------------------------------------------
__device__ __forceinline__ float softmin2_f(float p, float q) {
    float m = fminf(p, q);
    float t = __expf(-fabsf(p - q));
    return m - __logf(1.0f + t);
}

// trans with precomputed a and a^2:  c + (1-g)*softmin2(a^2, b^2)
__device__ __forceinline__ float trans_f(float a, float a2, float b) {
    float u = a * b;
    float g = 0.5f * (1.0f - u * __frsqrt_rn(u * u + 1e-9f));
    return 1.0f + (1.0f - g) * softmin2_f(a2, b * b);
}

__device__ __forceinline__ float softmin3_f(float a, float b, float c) {
    float m = fminf(a, fminf(b, c));
    float s = __expf(m - a) + __expf(m - b) + __expf(m - c);
    return m - __logf(s);
}

// ---------------------------------------------------------------------------
// Anti-diagonal wavefront DP. One workgroup per batch element, thread j owns
// column j. Three rotating LDS diagonals; x/y staged via async-to-LDS.
// ---------------------------------------------------------------------------
__global__ __launch_bounds__(TLEN) void softmsm_dp_kernel(
    const float* __restrict__ x, const float* __restrict__ y,
    float* __restrict__ ws) {
    __shared__ float xs[TLEN];
    __shared__ float ys[TLEN];
    __shared__ float buf[3][TLEN];

    const int j = threadIdx.x;
    const int b = blockIdx.x;

    const float* xb = x + b * TLEN;
    const float* yb = y + b * TLEN;

    // --- Stage x,y into LDS with gfx1250 async loads (ASYNCcnt path) ------
    {
        unsigned lx  = (unsigned)(uintptr_t)(&xs[j]);  // LDS byte address
        unsigned ly  = (unsigned)(uintptr_t)(&ys[j]);
        unsigned off = (unsigned)(j * 4);              // per-lane byte offset
        asm volatile(
            "global_load_async_to_lds_b32 %0, %2, %3 offset:0\n"
            "global_load_async_to_lds_b32 %1, %2, %4 offset:0\n"
            :
            : "v"(lx), "v"(ly), "v"(off), "s"(xb), "s"(yb)
            : "memory");
        asm volatile("s_wait_asynccnt 0" ::: "memory");
    }
    __syncthreads();

    // Loop-invariant pieces of the 'left' transition for column j:
    //   left(i,j) = trans(y_j, y_{j-1}, x_i); a = y_j - y_{j-1} is invariant.
    const float yj   = ys[j];
    const float yjm1 = (j > 0) ? ys[j - 1] : 0.0f;
    const float aL   = yj - yjm1;
    const float aL2  = aL * aL;

    float* prev2 = buf[0];
    float* prev  = buf[1];
    float* cur   = buf[2];

    float result = 0.0f;

    for (int d = 0; d < 2 * TLEN - 1; ++d) {
        const int i = d - j;  // row index handled by this thread on diagonal d
        if (i >= 0 && i < TLEN) {
            const float xi = xs[i];
            float v;
            if (d == 0) {
                // C[0][0] = (x0 - y0)^2
                float t = xi - yj;
                v = t * t;
            } else if (i == 0) {
                // row 0 cumulative: C[0][j] = C[0][j-1] + trans(y_j, y_{j-1}, x_0)
                v = prev[j - 1] + trans_f(aL, aL2, yj - xi);
            } else if (j == 0) {
                // col 0 cumulative: C[i][0] = C[i-1][0] + trans(x_i, x_{i-1}, y_0)
                const float aU = xi - xs[i - 1];
                v = prev[0] + trans_f(aU, aU * aU, xi - yj);
            } else {
                const float aU   = xi - xs[i - 1];
                const float diff = xi - yj;
                float dd = prev2[j - 1] + diff * diff;                 // diagonal
                float du = prev[j]      + trans_f(aU, aU * aU, diff);  // up
                float dl = prev[j - 1]  + trans_f(aL, aL2, yj - xi);   // left
                v = softmin3_f(dd, du, dl);
            }
            cur[j] = v;
            if (d == 2 * TLEN - 2) result = v;  // only thread j==TLEN-1 here
        }
        __syncthreads();
        float* t = prev2; prev2 = prev; prev = cur; cur = t;
    }

    if (j == TLEN - 1) ws[b] = result;  // C[T-1][U-1] for this batch element
}

// ---------------------------------------------------------------------------
// Mean over 64 costs via V_WMMA_F32_16X16X4_F32.
// A (16x4) holds the costs (cost index = M*4 + K):
//   lanes 0-15 : VGPR0 = K=0, VGPR1 = K=1 (M = lane)
//   lanes 16-31: VGPR0 = K=2, VGPR1 = K=3 (M = lane-16)
// B = all 1/64  =>  D[m][n] = rowsum(A,m)/64.  Lane L's 8 accumulators cover
// rows 0-7 (L<16) or 8-15 (L>=16); partner-lane shuffle gives the full mean.
// ---------------------------------------------------------------------------
typedef __attribute__((ext_vector_type(2))) float v2f;
typedef __attribute__((ext_vector_type(8))) float v8f;

__global__ __launch_bounds__(32) void reduce_mean_wmma_kernel(
    const float* __restrict__ ws, float* __restrict__ out) {
    const int lane = threadIdx.x;  // exactly one wave32, EXEC all ones
    const int m  = (lane < 16) ? lane : (lane - 16);
    const int k0 = (lane < 16) ? 0 : 2;

    v2f a;
    a.x = ws[m * 4 + k0];
    a.y = ws[m * 4 + k0 + 1];
    v2f bm;
    bm.x = 1.0f / 64.0f;
    bm.y = 1.0f / 64.0f;
    v8f c = {};
    c = __builtin_amdgcn_wmma_f32_16x16x4_f32(
        /*neg_a=*/false, a, /*neg_b=*/false, bm,
        /*c_mod=*/(short)0, c, /*reuse_a=*/false, /*reuse_b=*/false);

    float s = c[0] + c[1] + c[2] + c[3] + c[4] + c[5] + c[6] + c[7];
    float other = __shfl(s, lane ^ 16, 32);
    if (lane == 0) out[0] = s + other;  // mean over the batch
}

// ---------------------------------------------------------------------------
extern "C" void kernel_launch(void* const* d_in, const int* in_sizes, int n_in,
                              void* d_out, int out_size, void* d_ws, size_t ws_size,
                              hipStream_t stream) {
    const float* x = (const float*)d_in[0];  // (64, 1, 512) f32
    const float* y = (const float*)d_in[1];  // (64, 1, 512) f32
    float* ws = (float*)d_ws;                // 64 per-batch costs

    softmsm_dp_kernel<<<BATCH, TLEN, 0, stream>>>(x, y, ws);
    reduce_mean_wmma_kernel<<<1, 32, 0, stream>>>(ws, (float*)d_out);
}